// PointContrastiveLoss_30434138259690
// MI455X (gfx1250) — compile-verified
//
#include <hip/hip_runtime.h>
#include <math.h>

typedef __attribute__((ext_vector_type(2))) float v2f;
typedef __attribute__((ext_vector_type(8))) float v8f;

#define BB 2
#define NN 4096
#define DD 32
#define PP 1024
#define INV_T (1.0f / 0.07f)

__global__ void pcl_init_loss(float* out_loss) {
  if (threadIdx.x == 0 && blockIdx.x == 0) out_loss[0] = 0.0f;
}

// L2-normalize rows of a (nrows x 32) matrix.
__global__ void pcl_normalize_rows(const float* __restrict__ src,
                                   float* __restrict__ dst, int nrows) {
  int r = blockIdx.x * blockDim.x + threadIdx.x;
  if (r >= nrows) return;
  const float* s = src + (size_t)r * DD;
  float acc = 0.f;
#pragma unroll
  for (int i = 0; i < DD; ++i) { float v = s[i]; acc += v * v; }
  float inv = 1.0f / fmaxf(sqrtf(acc), 1e-12f);
  float* d = dst + (size_t)r * DD;
#pragma unroll
  for (int i = 0; i < DD; ++i) d[i] = s[i] * inv;
}

// qn[b,p,:] = normalize(q_feats[b, q_idx[b,p], :])
__global__ void pcl_gather_normalize_q(const float* __restrict__ qf,
                                       const int* __restrict__ q_idx,
                                       float* __restrict__ qn) {
  int i = blockIdx.x * blockDim.x + threadIdx.x;
  if (i >= BB * PP) return;
  int b = i / PP;
  int row = q_idx[i];
  const float* s = qf + ((size_t)b * NN + row) * DD;
  float acc = 0.f;
#pragma unroll
  for (int k = 0; k < DD; ++k) { float v = s[k]; acc += v * v; }
  float inv = 1.0f / fmaxf(sqrtf(acc), 1e-12f);
  float* d = qn + (size_t)i * DD;
#pragma unroll
  for (int k = 0; k < DD; ++k) d[k] = s[k] * inv;
}

// One workgroup (8 waves) per (b,p) row of the PxP logit matrix.
// WMMA f32 16x16x4 chained x8 over K=32; A = q replicated, B = 16 gathered
// t-rows as columns. Then fused log-softmax + loss accumulation.
__global__ void __launch_bounds__(256) pcl_fused(
    const float* __restrict__ qn, const float* __restrict__ tn,
    const int* __restrict__ pos_idx, const int* __restrict__ neg_idx,
    float* __restrict__ out_loss, float* __restrict__ out_logits) {
  __shared__ float sh_q[DD];
  __shared__ float sh_row[PP];
  __shared__ float sh_red[8];
  __shared__ float sh_scalar;

  int bp = blockIdx.x;  // 0 .. B*P-1
  int b = bp / PP;
  int p = bp - b * PP;
  int tid = threadIdx.x;
  int wave = tid >> 5;
  int lane = tid & 31;

  if (tid < DD) sh_q[tid] = qn[(size_t)bp * DD + tid];
  __syncthreads();

  const int* nrow = neg_idx + (size_t)bp * (PP - 1);
  int prow = pos_idx[bp];
  const float* tb = tn + (size_t)b * NN * DD;
  int koff = (lane < 16) ? 0 : 2;  // K-pair held by this lane half

  // 64 column-tiles of 16; 8 tiles per wave. EXEC is all-ones at every WMMA.
  for (int tc = wave; tc < PP / 16; tc += 8) {
    int col = tc * 16 + (lane & 15);
    int tr = (col == p) ? prow : nrow[col - (col > p ? 1 : 0)];
    const float* brow = tb + (size_t)tr * DD;
    v8f acc = {};
#pragma unroll
    for (int s = 0; s < 8; ++s) {
      v2f a, bv;
      // A (16x4 f32): lanes 0-15 = rows, hold k=4s,4s+1; lanes 16-31 hold
      // k=4s+2,4s+3. All rows identical (= q).
      a[0] = sh_q[4 * s + koff];
      a[1] = sh_q[4 * s + koff + 1];
      // B (4x16 f32): lane&15 = column j, same k-pair split.
      bv[0] = brow[4 * s + koff];
      bv[1] = brow[4 * s + koff + 1];
      acc = __builtin_amdgcn_wmma_f32_16x16x4_f32(
          /*neg_a=*/false, a, /*neg_b=*/false, bv,
          /*c_mod=*/(short)0, acc, /*reuse_a=*/false, /*reuse_b=*/false);
    }
    // All 16 output rows equal; VGPR0 of lanes 0-15 = row M=0, cols 0-15.
    if (lane < 16) sh_row[tc * 16 + lane] = acc[0];
  }
  __syncthreads();

  // Scale by 1/T, write logits row, block-wide max.
  float m = -3.4e38f;
  float* orow = out_logits + (size_t)bp * PP;
  for (int c = tid; c < PP; c += 256) {
    float v = sh_row[c] * INV_T;
    orow[c] = v;
    m = fmaxf(m, v);
  }
#pragma unroll
  for (int off = 16; off > 0; off >>= 1) m = fmaxf(m, __shfl_down(m, off, 32));
  if (lane == 0) sh_red[wave] = m;
  __syncthreads();
  if (tid == 0) {
    float mm = sh_red[0];
#pragma unroll
    for (int i = 1; i < 8; ++i) mm = fmaxf(mm, sh_red[i]);
    sh_scalar = mm;
  }
  __syncthreads();
  float rmax = sh_scalar;

  // Block-wide sum of exp.
  float ssum = 0.f;
  for (int c = tid; c < PP; c += 256) ssum += __expf(sh_row[c] * INV_T - rmax);
#pragma unroll
  for (int off = 16; off > 0; off >>= 1) ssum += __shfl_down(ssum, off, 32);
  if (lane == 0) sh_red[wave] = ssum;
  __syncthreads();
  if (tid == 0) {
    float tot = 0.f;
#pragma unroll
    for (int i = 0; i < 8; ++i) tot += sh_red[i];
    float lse = rmax + __logf(tot);
    float picked = sh_row[p] * INV_T - lse;  // label[p] == p (arange)
    atomicAdd(out_loss, -picked / (float)PP);
  }
}

extern "C" void kernel_launch(void* const* d_in, const int* in_sizes, int n_in,
                              void* d_out, int out_size, void* d_ws,
                              size_t ws_size, hipStream_t stream) {
  const float* q_feats = (const float*)d_in[0];
  const float* t_feats = (const float*)d_in[1];
  // d_in[2], d_in[3] (point clouds) and d_in[4] (labels == arange) unused.
  const int* q_idx = (const int*)d_in[5];
  const int* pos_idx = (const int*)d_in[6];
  const int* neg_idx = (const int*)d_in[7];
  float* out = (float*)d_out;  // [0] = loss, [1:] = batch_logits (B*P*P)

  float* tn = (float*)d_ws;                 // B*N*D normalized t  (1 MB)
  float* qn = tn + (size_t)BB * NN * DD;    // B*P*D gathered+normalized q

  pcl_init_loss<<<1, 1, 0, stream>>>(out);
  pcl_normalize_rows<<<(BB * NN + 255) / 256, 256, 0, stream>>>(t_feats, tn,
                                                                BB * NN);
  pcl_gather_normalize_q<<<(BB * PP + 255) / 256, 256, 0, stream>>>(q_feats,
                                                                    q_idx, qn);
  pcl_fused<<<BB * PP, 256, 0, stream>>>(qn, tn, pos_idx, neg_idx, out,
                                         out + 1);
}